// PWCN_POS_87393994539094
// MI455X (gfx1250) — compile-verified
//
#include <hip/hip_runtime.h>
#include <hip/hip_bf16.h>

// ---------------- problem constants ----------------
#define Bn   128
#define Tn   512
#define Vn   30000
#define En   300
#define Hn   300
#define Pn   3
#define C2H  600          // 2*H
#define Ep   304          // E padded to mult of 16
#define Hp   304          // H padded to mult of 16
#define Cp   608          // 2H padded to mult of 16
#define NT   19           // Hp/16 hidden-col tiles
#define K8G  38           // Ep/8 (== Hp/8) K-chunks for gate GEMMs
#define K8C  76           // Cp/8 K-chunks for conv GEMM
#define OTC  38           // Cp/16 out-channel tiles for conv

typedef __attribute__((ext_vector_type(2))) float v2f;
typedef __attribute__((ext_vector_type(4))) float v4f;
typedef __attribute__((ext_vector_type(8))) float v8f;

// f32 WMMA: D(16x16) = A(16x4) * B(4x16) + C   -- exact fp32 MACs on the matrix pipe
__device__ __forceinline__ v8f wmma_f32(v2f a, v2f b, v8f c) {
    return __builtin_amdgcn_wmma_f32_16x16x4_f32(
        /*neg_a=*/false, a, /*neg_b=*/false, b,
        /*c_mod=*/(short)0, c, /*reuse_a=*/false, /*reuse_b=*/false);
}

__device__ __forceinline__ float sigmoidf(float x) {
    return 1.0f / (1.0f + __expf(-x));
}

// ---------------- lengths (count nonzero tokens) ----------------
__global__ void k_lengths(const int* __restrict__ ti, const int* __restrict__ ai,
                          const int* __restrict__ li,
                          int* __restrict__ lt, int* __restrict__ la, int* __restrict__ ll) {
    int b = blockIdx.x;
    __shared__ int s0[256], s1[256], s2[256];
    int c0 = 0, c1 = 0, c2 = 0;
    for (int t = threadIdx.x; t < Tn; t += 256) {
        c0 += (ti[b * Tn + t] != 0);
        c1 += (ai[b * Tn + t] != 0);
        c2 += (li[b * Tn + t] != 0);
    }
    s0[threadIdx.x] = c0; s1[threadIdx.x] = c1; s2[threadIdx.x] = c2;
    __syncthreads();
    for (int off = 128; off > 0; off >>= 1) {
        if (threadIdx.x < off) {
            s0[threadIdx.x] += s0[threadIdx.x + off];
            s1[threadIdx.x] += s1[threadIdx.x + off];
            s2[threadIdx.x] += s2[threadIdx.x + off];
        }
        __syncthreads();
    }
    if (threadIdx.x == 0) { lt[b] = s0[0]; la[b] = s1[0]; ll[b] = s2[0]; }
}

// ---------------- fragment-packed gate weights ----------------
// dst[((g*NT + ntile)*K8G + k8)*128 + lane*4 + j] =
//   W[g*Hn + ntile*16 + (lane&15)][k8*8 + (j>>1)*4 + (lane>>4)*2 + (j&1)]   (0 if OOB)
// One b128 per lane then yields B fragments for two consecutive 16x16x4 WMMA K-steps.
__global__ void k_pack_gate_frag(const float* __restrict__ wih_f, const float* __restrict__ whh_f,
                                 const float* __restrict__ wih_b, const float* __restrict__ whh_b,
                                 float* __restrict__ fih_f, float* __restrict__ fhh_f,
                                 float* __restrict__ fih_b, float* __restrict__ fhh_b) {
    int sel = blockIdx.y;
    const float* src = (sel == 0) ? wih_f : (sel == 1) ? whh_f : (sel == 2) ? wih_b : whh_b;
    float*       dst = (sel == 0) ? fih_f : (sel == 1) ? fhh_f : (sel == 2) ? fih_b : fhh_b;
    int blk = blockIdx.x;                 // ((g*NT + ntile)*K8G + k8)
    int k8 = blk % K8G;
    int r  = blk / K8G;
    int ntile = r % NT;
    int g     = r / NT;
    int tid  = threadIdx.x;               // 0..127
    int lane = tid >> 2, j = tid & 3;
    int row = ntile * 16 + (lane & 15);
    int K   = k8 * 8 + ((j >> 1) * 4) + ((lane >> 4) * 2) + (j & 1);
    float v = (row < Hn && K < En) ? src[(size_t)(g * Hn + row) * En + K] : 0.0f;
    dst[(size_t)blk * 128 + tid] = v;
}

// fragment-packed conv weights (A-matrix): one b128 per lane covers 2 K-steps.
// dst[((ot*3 + tap)*K8C + k8)*128 + lane*4 + j] = conv_w[o][i][tap],
//   o = ot*16 + (lane&15),  i = k8*8 + (j>>1)*4 + (lane>>4)*2 + (j&1)
__global__ void k_pack_conv_frag(const float* __restrict__ cw, float* __restrict__ dst) {
    int blk = blockIdx.x;                 // ((ot*3 + tap)*K8C + k8)
    int k8 = blk % K8C;
    int r  = blk / K8C;
    int tap = r % 3;
    int ot  = r / 3;
    int tid  = threadIdx.x;
    int lane = tid >> 2, j = tid & 3;
    int o = ot * 16 + (lane & 15);
    int i = k8 * 8 + ((j >> 1) * 4) + ((lane >> 4) * 2) + (j & 1);
    float v = (o < C2H && i < C2H) ? cw[((size_t)o * C2H + i) * 3 + tap] : 0.0f;
    dst[(size_t)blk * 128 + tid] = v;
}

// ---------------- embedding gather into padded (B,T,Ep) ----------------
__global__ void k_embed(const int* __restrict__ ti, const float* __restrict__ embd,
                        float* __restrict__ emb) {
    int t = blockIdx.x, b = blockIdx.y;
    int tok = ti[b * Tn + t];
    float* dst = emb + ((size_t)b * Tn + t) * Ep;
    const float* src = embd + (size_t)tok * En;
    for (int e = threadIdx.x; e < Ep; e += blockDim.x)
        dst[e] = (e < En) ? src[e] : 0.0f;
}

// ---------------- one LSTM time step (both directions), WMMA f32 ----------------
// grid: (B/32, NT, 2 dirs); block: 32.
// Each wave: 2 batch tiles x 4 gates. Weights come from the fragment-packed
// layout: 4x b128 + 4x b64 loads -> 16 WMMAs per K8 chunk (0.5 loads/WMMA).
__global__ void k_lstm_step(int t,
                            const float* __restrict__ emb,
                            const float* __restrict__ fih_f, const float* __restrict__ fhh_f,
                            const float* __restrict__ fih_b, const float* __restrict__ fhh_b,
                            const float* __restrict__ bias_f, const float* __restrict__ bias_b,
                            const int* __restrict__ txt_len,
                            const float* __restrict__ zrow,
                            float* __restrict__ hbuf,   // [parity][dir][B][Hp]
                            float* __restrict__ cst,    // [dir][B][Hp]
                            float* __restrict__ hseq_f, float* __restrict__ hseq_b) {
    const int lane = threadIdx.x;
    const int mtp = blockIdx.x;       // batch tile-pair (covers 32 batch rows)
    const int nt  = blockIdx.y;       // hidden-col tile
    const int dir = blockIdx.z;
    const int l15 = lane & 15;
    const int hk  = (lane >> 4) * 2;  // K sub-offset for f32 fragments

    const int b0 = mtp * 32 + l15;    // A rows (batch) for the two M tiles
    const int b1 = b0 + 16;

    const float* fih  = dir ? fih_b : fih_f;
    const float* fhh  = dir ? fhh_b : fhh_f;
    const float* bias = dir ? bias_b : bias_f;

    // per-lane x row pointers; masked rows point at a zeroed scratch row
    const int tl0 = txt_len[b0];
    const int tl1 = txt_len[b1];
    const float* x0;
    const float* x1;
    if (dir == 0) {
        x0 = emb + ((size_t)b0 * Tn + t) * Ep;
        x1 = emb + ((size_t)b1 * Tn + t) * Ep;
    } else {
        x0 = (t < tl0) ? (emb + ((size_t)b0 * Tn + (tl0 - 1 - t)) * Ep) : zrow;
        x1 = (t < tl1) ? (emb + ((size_t)b1 * Tn + (tl1 - 1 - t)) * Ep) : zrow;
    }
    // prefetch next step's x rows (global_prefetch_b8) to warm caches for launch t+1
    if (t + 1 < Tn) {
        const float* p0;
        const float* p1;
        if (dir == 0) {
            p0 = emb + ((size_t)b0 * Tn + (t + 1)) * Ep;
            p1 = emb + ((size_t)b1 * Tn + (t + 1)) * Ep;
        } else {
            p0 = (t + 1 < tl0) ? (emb + ((size_t)b0 * Tn + (tl0 - 2 - t)) * Ep) : zrow;
            p1 = (t + 1 < tl1) ? (emb + ((size_t)b1 * Tn + (tl1 - 2 - t)) * Ep) : zrow;
        }
        __builtin_prefetch(p0, 0, 3);
        __builtin_prefetch(p1, 0, 3);
    }
    const float* h0 = hbuf + ((((size_t)(t & 1)) * 2 + dir) * Bn + b0) * Hp;
    const float* h1 = hbuf + ((((size_t)(t & 1)) * 2 + dir) * Bn + b1) * Hp;

    // fragment-packed weight pointers (per gate, this nt tile)
    const size_t lb = (size_t)lane * 4;
    const float* wiP0 = fih + (((size_t)(0 * NT + nt)) * K8G) * 128 + lb;
    const float* wiP1 = fih + (((size_t)(1 * NT + nt)) * K8G) * 128 + lb;
    const float* wiP2 = fih + (((size_t)(2 * NT + nt)) * K8G) * 128 + lb;
    const float* wiP3 = fih + (((size_t)(3 * NT + nt)) * K8G) * 128 + lb;
    const float* whP0 = fhh + (((size_t)(0 * NT + nt)) * K8G) * 128 + lb;
    const float* whP1 = fhh + (((size_t)(1 * NT + nt)) * K8G) * 128 + lb;
    const float* whP2 = fhh + (((size_t)(2 * NT + nt)) * K8G) * 128 + lb;
    const float* whP3 = fhh + (((size_t)(3 * NT + nt)) * K8G) * 128 + lb;

    v8f acc[2][4];
#pragma unroll
    for (int mi = 0; mi < 2; mi++)
#pragma unroll
        for (int g = 0; g < 4; g++)
            acc[mi][g] = (v8f){};

    // part 1: x_t @ Wih^T  (K8G chunks of K=8 -> 16 WMMAs each)
    for (int k8 = 0; k8 < K8G; k8++) {
        const int koff = k8 * 8;
        v4f G0 = *(const v4f*)(wiP0 + (size_t)k8 * 128);
        v4f G1 = *(const v4f*)(wiP1 + (size_t)k8 * 128);
        v4f G2 = *(const v4f*)(wiP2 + (size_t)k8 * 128);
        v4f G3 = *(const v4f*)(wiP3 + (size_t)k8 * 128);
        v2f a0l = *(const v2f*)(x0 + koff + hk);
        v2f a1l = *(const v2f*)(x1 + koff + hk);
        v2f a0h = *(const v2f*)(x0 + koff + 4 + hk);
        v2f a1h = *(const v2f*)(x1 + koff + 4 + hk);
        v2f g0l = {G0.x, G0.y}, g0h = {G0.z, G0.w};
        v2f g1l = {G1.x, G1.y}, g1h = {G1.z, G1.w};
        v2f g2l = {G2.x, G2.y}, g2h = {G2.z, G2.w};
        v2f g3l = {G3.x, G3.y}, g3h = {G3.z, G3.w};
        acc[0][0] = wmma_f32(a0l, g0l, acc[0][0]);
        acc[0][1] = wmma_f32(a0l, g1l, acc[0][1]);
        acc[0][2] = wmma_f32(a0l, g2l, acc[0][2]);
        acc[0][3] = wmma_f32(a0l, g3l, acc[0][3]);
        acc[1][0] = wmma_f32(a1l, g0l, acc[1][0]);
        acc[1][1] = wmma_f32(a1l, g1l, acc[1][1]);
        acc[1][2] = wmma_f32(a1l, g2l, acc[1][2]);
        acc[1][3] = wmma_f32(a1l, g3l, acc[1][3]);
        acc[0][0] = wmma_f32(a0h, g0h, acc[0][0]);
        acc[0][1] = wmma_f32(a0h, g1h, acc[0][1]);
        acc[0][2] = wmma_f32(a0h, g2h, acc[0][2]);
        acc[0][3] = wmma_f32(a0h, g3h, acc[0][3]);
        acc[1][0] = wmma_f32(a1h, g0h, acc[1][0]);
        acc[1][1] = wmma_f32(a1h, g1h, acc[1][1]);
        acc[1][2] = wmma_f32(a1h, g2h, acc[1][2]);
        acc[1][3] = wmma_f32(a1h, g3h, acc[1][3]);
    }
    // part 2: h_{t-1} @ Whh^T
    for (int k8 = 0; k8 < K8G; k8++) {
        const int koff = k8 * 8;
        v4f G0 = *(const v4f*)(whP0 + (size_t)k8 * 128);
        v4f G1 = *(const v4f*)(whP1 + (size_t)k8 * 128);
        v4f G2 = *(const v4f*)(whP2 + (size_t)k8 * 128);
        v4f G3 = *(const v4f*)(whP3 + (size_t)k8 * 128);
        v2f a0l = *(const v2f*)(h0 + koff + hk);
        v2f a1l = *(const v2f*)(h1 + koff + hk);
        v2f a0h = *(const v2f*)(h0 + koff + 4 + hk);
        v2f a1h = *(const v2f*)(h1 + koff + 4 + hk);
        v2f g0l = {G0.x, G0.y}, g0h = {G0.z, G0.w};
        v2f g1l = {G1.x, G1.y}, g1h = {G1.z, G1.w};
        v2f g2l = {G2.x, G2.y}, g2h = {G2.z, G2.w};
        v2f g3l = {G3.x, G3.y}, g3h = {G3.z, G3.w};
        acc[0][0] = wmma_f32(a0l, g0l, acc[0][0]);
        acc[0][1] = wmma_f32(a0l, g1l, acc[0][1]);
        acc[0][2] = wmma_f32(a0l, g2l, acc[0][2]);
        acc[0][3] = wmma_f32(a0l, g3l, acc[0][3]);
        acc[1][0] = wmma_f32(a1l, g0l, acc[1][0]);
        acc[1][1] = wmma_f32(a1l, g1l, acc[1][1]);
        acc[1][2] = wmma_f32(a1l, g2l, acc[1][2]);
        acc[1][3] = wmma_f32(a1l, g3l, acc[1][3]);
        acc[0][0] = wmma_f32(a0h, g0h, acc[0][0]);
        acc[0][1] = wmma_f32(a0h, g1h, acc[0][1]);
        acc[0][2] = wmma_f32(a0h, g2h, acc[0][2]);
        acc[0][3] = wmma_f32(a0h, g3h, acc[0][3]);
        acc[1][0] = wmma_f32(a1h, g0h, acc[1][0]);
        acc[1][1] = wmma_f32(a1h, g1h, acc[1][1]);
        acc[1][2] = wmma_f32(a1h, g2h, acc[1][2]);
        acc[1][3] = wmma_f32(a1h, g3h, acc[1][3]);
    }

    // epilogue: C/D layout -> m = r + (lane>=16 ? 8 : 0), n = lane&15
    float* hout = hbuf + (((size_t)((t + 1) & 1)) * 2 + dir) * Bn * Hp;
    float* hseq = dir ? hseq_b : hseq_f;
    const int hk8 = (lane >> 4) * 8;
    const int hc = nt * 16 + l15;
    const float bi = (hc < Hn) ? bias[0 * Hn + hc] : 0.0f;
    const float bf = (hc < Hn) ? bias[1 * Hn + hc] : 0.0f;
    const float bg = (hc < Hn) ? bias[2 * Hn + hc] : 0.0f;
    const float bo = (hc < Hn) ? bias[3 * Hn + hc] : 0.0f;
#pragma unroll
    for (int mi = 0; mi < 2; mi++) {
#pragma unroll
        for (int r = 0; r < 8; r++) {
            int bb = mtp * 32 + mi * 16 + r + hk8;    // batch index of this C element
            float gi = sigmoidf(acc[mi][0][r] + bi);
            float gf = sigmoidf(acc[mi][1][r] + bf);
            float gg = tanhf(acc[mi][2][r] + bg);
            float go = sigmoidf(acc[mi][3][r] + bo);
            size_t cidx = ((size_t)dir * Bn + bb) * Hp + hc;
            float c_new = gf * cst[cidx] + gi * gg;
            cst[cidx] = c_new;
            float h_new = go * tanhf(c_new);
            hout[(size_t)bb * Hp + hc] = h_new;
            hseq[((size_t)bb * Tn + t) * Hp + hc] = h_new;
        }
    }
}

// ---------------- mask + reverse-bwd + proximity weight + concat ----------------
__global__ void k_mix(const float* __restrict__ hf, const float* __restrict__ hb,
                      const int* __restrict__ lt, const int* __restrict__ la,
                      const int* __restrict__ ll, float* __restrict__ hcat) {
    int t = blockIdx.x, b = blockIdx.y;
    int tl = lt[b], al = la[b], L = ll[b];
    float ctx = (float)(tl - al);
    float j = (float)t;
    float a0 = (float)L;
    float a1 = (float)(L + al - 1);
    float w;
    if (j < a0)             w = 1.0f - (a0 - j) / ctx;
    else if (j <= a1)       w = 0.0f;
    else if (j < (float)tl) w = 1.0f - (j - a1) / ctx;
    else                    w = 0.0f;
    bool in = (t < tl);
    float* dst = hcat + ((size_t)b * Tn + t) * Cp;
    const float* hfrow = hf + ((size_t)b * Tn + t) * Hp;
    const float* hbrow = in ? (hb + ((size_t)b * Tn + (tl - 1 - t)) * Hp) : nullptr;
    for (int c = threadIdx.x; c < Cp; c += blockDim.x) {
        float v = 0.0f;
        if (in) {
            if (c < Hn)          v = hfrow[c];
            else if (c < 2 * Hn) v = hbrow[c - Hn];
        }
        dst[c] = w * v;
    }
}

// ---------------- conv1d(k=3, same) + ReLU + max-over-T, fused, WMMA f32 ----------------
// grid: (Cp/32, T/32, B); block: 32. 2x2 tile blocking; A from fragment-packed
// weights (b128), B rows from hcat (b64): 6 loads -> 8 WMMAs per K8 chunk.
__global__ void k_conv_pool(const float* __restrict__ hcat, const float* __restrict__ cwfrag,
                            const float* __restrict__ cb, const float* __restrict__ zrow,
                            float* __restrict__ pool) {
    const int lane = threadIdx.x;
    const int otp = blockIdx.x;       // out-channel tile pair
    const int ttp = blockIdx.y;       // time tile pair
    const int b   = blockIdx.z;
    const int l15 = lane & 15;
    const int hk  = (lane >> 4) * 2;
    const int n0 = ttp * 32 + l15;    // B cols (time)
    const int n1 = n0 + 16;
    const size_t lb = (size_t)lane * 4;

    v8f acc[2][2];
#pragma unroll
    for (int io = 0; io < 2; io++)
#pragma unroll
        for (int it = 0; it < 2; it++)
            acc[io][it] = (v8f){};

#pragma unroll
    for (int tap = 0; tap < 3; tap++) {
        const float* A0 = cwfrag + (((size_t)(otp * 2 + 0) * 3 + tap) * K8C) * 128 + lb;
        const float* A1 = cwfrag + (((size_t)(otp * 2 + 1) * 3 + tap) * K8C) * 128 + lb;
        int tn0 = n0 + tap - 1;
        int tn1 = n1 + tap - 1;
        const float* b0row = (tn0 >= 0 && tn0 < Tn) ? (hcat + ((size_t)b * Tn + tn0) * Cp) : zrow;
        const float* b1row = (tn1 >= 0 && tn1 < Tn) ? (hcat + ((size_t)b * Tn + tn1) * Cp) : zrow;
        for (int k8 = 0; k8 < K8C; k8++) {
            const int koff = k8 * 8;
            v4f Av0 = *(const v4f*)(A0 + (size_t)k8 * 128);
            v4f Av1 = *(const v4f*)(A1 + (size_t)k8 * 128);
            v2f w0l = *(const v2f*)(b0row + koff + hk);
            v2f w1l = *(const v2f*)(b1row + koff + hk);
            v2f w0h = *(const v2f*)(b0row + koff + 4 + hk);
            v2f w1h = *(const v2f*)(b1row + koff + 4 + hk);
            v2f a0l = {Av0.x, Av0.y}, a0h = {Av0.z, Av0.w};
            v2f a1l = {Av1.x, Av1.y}, a1h = {Av1.z, Av1.w};
            acc[0][0] = wmma_f32(a0l, w0l, acc[0][0]);
            acc[0][1] = wmma_f32(a0l, w1l, acc[0][1]);
            acc[1][0] = wmma_f32(a1l, w0l, acc[1][0]);
            acc[1][1] = wmma_f32(a1l, w1l, acc[1][1]);
            acc[0][0] = wmma_f32(a0h, w0h, acc[0][0]);
            acc[0][1] = wmma_f32(a0h, w1h, acc[0][1]);
            acc[1][0] = wmma_f32(a1h, w0h, acc[1][0]);
            acc[1][1] = wmma_f32(a1h, w1h, acc[1][1]);
        }
    }

    const int hk8 = (lane >> 4) * 8;
#pragma unroll
    for (int io = 0; io < 2; io++) {
#pragma unroll
        for (int it = 0; it < 2; it++) {
#pragma unroll
            for (int r = 0; r < 8; r++) {
                int o = otp * 32 + io * 16 + r + hk8;
                float z = acc[io][it][r] + ((o < C2H) ? cb[o] : 0.0f);
                z = fmaxf(z, 0.0f);
                // max over the 16 time lanes (stay within each 16-lane half)
                for (int m = 1; m <= 8; m <<= 1)
                    z = fmaxf(z, __shfl_xor(z, m, 32));
                if (l15 == 0)
                    atomicMax((int*)(pool + (size_t)b * Cp + o), __float_as_int(z));
            }
        }
    }
}

// ---------------- FC 600 -> 3 ----------------
__global__ void k_fc(const float* __restrict__ pool, const float* __restrict__ fcw,
                     const float* __restrict__ fcb, float* __restrict__ out) {
    int p = blockIdx.x, b = blockIdx.y;
    float s = 0.0f;
    for (int o = threadIdx.x; o < C2H; o += 32)
        s += pool[(size_t)b * Cp + o] * fcw[p * C2H + o];
    for (int m = 16; m > 0; m >>= 1)
        s += __shfl_xor(s, m, 32);
    if (threadIdx.x == 0) out[b * Pn + p] = s + fcb[p];
}

extern "C" void kernel_launch(void* const* d_in, const int* in_sizes, int n_in,
                              void* d_out, int out_size, void* d_ws, size_t ws_size,
                              hipStream_t stream) {
    (void)in_sizes; (void)n_in; (void)out_size; (void)ws_size;
    const int*   ti    = (const int*)d_in[0];
    const int*   ai    = (const int*)d_in[1];
    const int*   li    = (const int*)d_in[2];
    const float* embd  = (const float*)d_in[3];
    const float* wih_f = (const float*)d_in[4];
    const float* whh_f = (const float*)d_in[5];
    const float* b_f   = (const float*)d_in[6];
    const float* wih_b = (const float*)d_in[7];
    const float* whh_b = (const float*)d_in[8];
    const float* b_b   = (const float*)d_in[9];
    const float* cw    = (const float*)d_in[10];
    const float* cb    = (const float*)d_in[11];
    const float* fcw   = (const float*)d_in[12];
    const float* fcb   = (const float*)d_in[13];
    float* out = (float*)d_out;

    float* ws = (float*)d_ws;
    size_t off = 0;
    int* lt = (int*)(ws + off);                 // 3*B ints in 512-float slot
    int* la = lt + Bn;
    int* ll = la + Bn;
    off += 512;
    float* emb    = ws + off; off += (size_t)Bn * Tn * Ep;
    float* hf     = ws + off; off += (size_t)Bn * Tn * Hp;
    float* hb     = ws + off; off += (size_t)Bn * Tn * Hp;
    float* hcat   = ws + off; off += (size_t)Bn * Tn * Cp;
    float* fih_f  = ws + off; off += (size_t)4 * NT * K8G * 128;   // frag-packed Wih fwd
    float* fhh_f  = ws + off; off += (size_t)4 * NT * K8G * 128;
    float* fih_b  = ws + off; off += (size_t)4 * NT * K8G * 128;
    float* fhh_b  = ws + off; off += (size_t)4 * NT * K8G * 128;
    float* cwfrag = ws + off; off += (size_t)OTC * 3 * K8C * 128;  // frag-packed conv W
    float* hbuf   = ws + off; off += (size_t)2 * 2 * Bn * Hp;      // [parity][dir][B][Hp]
    float* cst    = ws + off; off += (size_t)2 * Bn * Hp;          // [dir][B][Hp]
    float* pool   = ws + off; off += (size_t)Bn * Cp;
    float* zrow   = ws + off; off += (size_t)Cp;                   // shared all-zero row

    // zero-init recurrent state (parity-0 h buffers), cell state, pool, zero-row
    hipMemsetAsync(hbuf, 0, (size_t)2 * Bn * Hp * sizeof(float), stream);
    hipMemsetAsync(cst,  0, (size_t)2 * Bn * Hp * sizeof(float), stream);
    hipMemsetAsync(pool, 0, (size_t)Bn * Cp * sizeof(float), stream);
    hipMemsetAsync(zrow, 0, (size_t)Cp * sizeof(float), stream);

    k_lengths<<<Bn, 256, 0, stream>>>(ti, ai, li, lt, la, ll);
    k_pack_gate_frag<<<dim3(4 * NT * K8G, 4), 128, 0, stream>>>(
        wih_f, whh_f, wih_b, whh_b, fih_f, fhh_f, fih_b, fhh_b);
    k_pack_conv_frag<<<dim3(OTC * 3 * K8C), 128, 0, stream>>>(cw, cwfrag);
    k_embed<<<dim3(Tn, Bn), 128, 0, stream>>>(ti, embd, emb);

    for (int t = 0; t < Tn; t++) {
        k_lstm_step<<<dim3(Bn / 32, NT, 2), 32, 0, stream>>>(
            t, emb, fih_f, fhh_f, fih_b, fhh_b, b_f, b_b, lt, zrow, hbuf, cst, hf, hb);
    }

    k_mix<<<dim3(Tn, Bn), 256, 0, stream>>>(hf, hb, lt, la, ll, hcat);
    k_conv_pool<<<dim3(Cp / 32, Tn / 32, Bn), 32, 0, stream>>>(hcat, cwfrag, cb, zrow, pool);
    k_fc<<<dim3(Pn, Bn), 32, 0, stream>>>(pool, fcw, fcb, out);
}